// NewLoss_14362370638221
// MI455X (gfx1250) — compile-verified
//
#include <hip/hip_runtime.h>
#include <cstdint>
#include <cstddef>

#define G      32
#define VOLN   (G * G * G)          // 32768 floats = 128 KB
#define NV     778
#define BLOCK  256
#define NWAVES (BLOCK / 32)

__device__ __forceinline__ float wave_sum(float v) {
    v += __shfl_xor(v, 16, 32);
    v += __shfl_xor(v, 8, 32);
    v += __shfl_xor(v, 4, 32);
    v += __shfl_xor(v, 2, 32);
    v += __shfl_xor(v, 1, 32);
    return v;
}
__device__ __forceinline__ float wave_min(float v) {
    v = fminf(v, __shfl_xor(v, 16, 32));
    v = fminf(v, __shfl_xor(v, 8, 32));
    v = fminf(v, __shfl_xor(v, 4, 32));
    v = fminf(v, __shfl_xor(v, 2, 32));
    v = fminf(v, __shfl_xor(v, 1, 32));
    return v;
}
__device__ __forceinline__ float wave_max(float v) {
    v = fmaxf(v, __shfl_xor(v, 16, 32));
    v = fmaxf(v, __shfl_xor(v, 8, 32));
    v = fmaxf(v, __shfl_xor(v, 4, 32));
    v = fmaxf(v, __shfl_xor(v, 2, 32));
    v = fmaxf(v, __shfl_xor(v, 1, 32));
    return v;
}

// Trilinear sample matching jax reference: clamped floor, f = c - clamped_floor(c)
__device__ __forceinline__ float tri_sample(const float* sv, float cx, float cy, float cz) {
    const float gm1 = (float)(G - 1);
    float ix = (cx + 1.0f) * 0.5f * gm1;
    float iy = (cy + 1.0f) * 0.5f * gm1;
    float iz = (cz + 1.0f) * 0.5f * gm1;
    float fx0 = fminf(fmaxf(floorf(ix), 0.0f), gm1);
    float fy0 = fminf(fmaxf(floorf(iy), 0.0f), gm1);
    float fz0 = fminf(fmaxf(floorf(iz), 0.0f), gm1);
    float fx = ix - fx0, fy = iy - fy0, fz = iz - fz0;
    int x0 = (int)fx0, y0 = (int)fy0, z0 = (int)fz0;
    int x1 = min(x0 + 1, G - 1), y1 = min(y0 + 1, G - 1), z1 = min(z0 + 1, G - 1);

    float c000 = sv[(z0 * G + y0) * G + x0];
    float c001 = sv[(z0 * G + y0) * G + x1];
    float c010 = sv[(z0 * G + y1) * G + x0];
    float c011 = sv[(z0 * G + y1) * G + x1];
    float c100 = sv[(z1 * G + y0) * G + x0];
    float c101 = sv[(z1 * G + y0) * G + x1];
    float c110 = sv[(z1 * G + y1) * G + x0];
    float c111 = sv[(z1 * G + y1) * G + x1];

    float wx0 = 1.0f - fx, wy0 = 1.0f - fy, wz0 = 1.0f - fz;
    return ((c000 * wx0 + c001 * fx) * wy0 + (c010 * wx0 + c011 * fx) * fy) * wz0 +
           ((c100 * wx0 + c101 * fx) * wy0 + (c110 * wx0 + c111 * fx) * fy) * fz;
}

__global__ __launch_bounds__(BLOCK) void hand_loss_kernel(
    const float* __restrict__ verts,   // [bs, 2, NV, 3]
    const float* __restrict__ phiR,    // [bs, G, G, G]
    const float* __restrict__ phiL,    // [bs, G, G, G]
    float* __restrict__ out)           // [bs]
{
    __shared__ __align__(16) float vol[VOLN];     // 128 KB volume buffer (reused)
    __shared__ float s_red[NWAVES][12];
    __shared__ float s_param[8];                  // center[2][3], scale[2]

    const int b    = blockIdx.x;
    const int tid  = threadIdx.x;
    const int wave = tid >> 5;
    const int lane = tid & 31;

    const float* vb = verts + (size_t)b * (2 * NV * 3);

    // ---------------- Phase 1: per-hand bounding boxes ----------------
    float mn[6], mx[6];
#pragma unroll
    for (int k = 0; k < 6; ++k) { mn[k] = 3.0e38f; mx[k] = -3.0e38f; }
    for (int v = tid; v < NV; v += BLOCK) {
#pragma unroll
        for (int h = 0; h < 2; ++h) {
            const float* p = vb + (size_t)(h * NV + v) * 3;
#pragma unroll
            for (int d = 0; d < 3; ++d) {
                float val = p[d];
                mn[h * 3 + d] = fminf(mn[h * 3 + d], val);
                mx[h * 3 + d] = fmaxf(mx[h * 3 + d], val);
            }
        }
    }
#pragma unroll
    for (int k = 0; k < 6; ++k) { mn[k] = wave_min(mn[k]); mx[k] = wave_max(mx[k]); }
    if (lane == 0) {
#pragma unroll
        for (int k = 0; k < 6; ++k) { s_red[wave][k] = mn[k]; s_red[wave][6 + k] = mx[k]; }
    }
    __syncthreads();
    if (tid == 0) {
#pragma unroll
        for (int h = 0; h < 2; ++h) {
            float ext = -3.0e38f;
#pragma unroll
            for (int d = 0; d < 3; ++d) {
                float lo = s_red[0][h * 3 + d];
                float hi = s_red[0][6 + h * 3 + d];
                for (int w = 1; w < NWAVES; ++w) {
                    lo = fminf(lo, s_red[w][h * 3 + d]);
                    hi = fmaxf(hi, s_red[w][6 + h * 3 + d]);
                }
                s_param[h * 3 + d] = 0.5f * (lo + hi);
                ext = fmaxf(ext, hi - lo);
            }
            s_param[6 + h] = 1.1f * 0.5f * ext;   // (1 + SCALE_FACTOR) * 0.5 * extent
        }
    }
    __syncthreads();

    const float c0x = s_param[0], c0y = s_param[1], c0z = s_param[2];
    const float c1x = s_param[3], c1y = s_param[4], c1z = s_param[5];
    const float sc0 = s_param[6], sc1 = s_param[7];

    const unsigned lds_base = (unsigned)(size_t)(void*)&vol[0];
    float acc = 0.0f;

    // -------- Pass A: phi_right[b] sampled at left-hand verts (hand-0 box) --------
    {
        const float* src = phiR + (size_t)b * VOLN;
        unsigned off = (unsigned)tid * 16u;
#pragma unroll
        for (int i = 0; i < (VOLN * 4) / (BLOCK * 16); ++i) {   // 32 issues, 128 KB total
            asm volatile("global_load_async_to_lds_b128 %0, %1, %2"
                         :: "v"(lds_base + off), "v"(off), "s"(src)
                         : "memory");
            off += BLOCK * 16;
        }
        asm volatile("s_wait_asynccnt 0" ::: "memory");
        __syncthreads();
        for (int v = tid; v < NV; v += BLOCK) {
            const float* p = vb + (size_t)(1 * NV + v) * 3;     // left_vert = vertices[:,1]
            acc += tri_sample(vol, (p[0] - c0x) / sc0, (p[1] - c0y) / sc0, (p[2] - c0z) / sc0);
        }
        __syncthreads();   // everyone done reading vol before it is overwritten
    }

    // -------- Pass B: phi_left[b] sampled at right-hand verts (hand-1 box) --------
    {
        const float* src = phiL + (size_t)b * VOLN;
        unsigned off = (unsigned)tid * 16u;
#pragma unroll
        for (int i = 0; i < (VOLN * 4) / (BLOCK * 16); ++i) {
            asm volatile("global_load_async_to_lds_b128 %0, %1, %2"
                         :: "v"(lds_base + off), "v"(off), "s"(src)
                         : "memory");
            off += BLOCK * 16;
        }
        asm volatile("s_wait_asynccnt 0" ::: "memory");
        __syncthreads();
        for (int v = tid; v < NV; v += BLOCK) {
            const float* p = vb + (size_t)(0 * NV + v) * 3;     // right_vert = vertices[:,0]
            acc += tri_sample(vol, (p[0] - c1x) / sc1, (p[1] - c1y) / sc1, (p[2] - c1z) / sc1);
        }
    }

    // ---------------- Final reduction: loss[b] ----------------
    acc *= 0.25f;                       // / num_hand**2, applied to both terms
    acc = wave_sum(acc);
    __syncthreads();                    // s_red reuse safety
    if (lane == 0) s_red[wave][0] = acc;
    __syncthreads();
    if (tid == 0) {
        float t = 0.0f;
        for (int w = 0; w < NWAVES; ++w) t += s_red[w][0];
        out[b] = t;
    }
}

extern "C" void kernel_launch(void* const* d_in, const int* in_sizes, int n_in,
                              void* d_out, int out_size, void* d_ws, size_t ws_size,
                              hipStream_t stream) {
    const float* verts = (const float*)d_in[0];   // [bs, 2, 778, 3]
    const float* phiR  = (const float*)d_in[1];   // [bs, 32, 32, 32]
    const float* phiL  = (const float*)d_in[2];   // [bs, 32, 32, 32]
    float* out = (float*)d_out;                   // [bs]

    const int bs = in_sizes[0] / (2 * NV * 3);
    hand_loss_kernel<<<bs, BLOCK, 0, stream>>>(verts, phiR, phiL, out);
}